// CGCN_2688649527441
// MI455X (gfx1250) — compile-verified
//
#include <hip/hip_runtime.h>
#include <hip/hip_bf16.h>

#define NUM_USER 100000
#define NUM_ITEM 50000
#define DIM_C 64
#define DIM_FEAT 384
#define NUM_EDGES 1000000
#define N_NODES (NUM_USER + NUM_ITEM)

typedef __attribute__((ext_vector_type(2))) float v2f;
typedef __attribute__((ext_vector_type(8))) float v8f;

// ---- order-preserving float <-> uint key for f32 atomic max ----
__device__ __forceinline__ unsigned fkey(float f) {
    unsigned u = __float_as_uint(f);
    return (u & 0x80000000u) ? ~u : (u | 0x80000000u);
}
__device__ __forceinline__ float ikey(unsigned k) {
    return __uint_as_float((k & 0x80000000u) ? (k & 0x7fffffffu) : ~k);
}
#define KEY_NEG_INF 0x007fffffu  // fkey(-inf)

__device__ __forceinline__ float wave_reduce_sum(float v) {
#pragma unroll
    for (int o = 16; o > 0; o >>= 1) v += __shfl_xor(v, o, 32);
    return v;
}

// ---------------------------------------------------------------------------
// feats = leaky_relu(features @ W.T + b)  via V_WMMA_F32_16X16X4_F32
// One wave computes a 16x64 output tile (4 N-tiles), looping K in steps of 4.
// ---------------------------------------------------------------------------
__global__ __launch_bounds__(256) void k_gemm_feats(
    const float* __restrict__ F, const float* __restrict__ W,
    const float* __restrict__ bias, float* __restrict__ out) {
    const int wavesPerBlk = blockDim.x >> 5;
    const int tile = blockIdx.x * wavesPerBlk + (threadIdx.x >> 5);
    const int lane = threadIdx.x & 31;
    if (tile >= NUM_ITEM / 16) return;
    const int rowBase = tile * 16;
    const int hl = lane >> 4;   // half-wave select
    const int l  = lane & 15;

    v8f acc[4] = {v8f{}, v8f{}, v8f{}, v8f{}};

    const float* arow = F + (size_t)(rowBase + l) * DIM_FEAT;
    const float* w0 = W + (size_t)(0 * 16 + l) * DIM_FEAT;
    const float* w1 = W + (size_t)(1 * 16 + l) * DIM_FEAT;
    const float* w2 = W + (size_t)(2 * 16 + l) * DIM_FEAT;
    const float* w3 = W + (size_t)(3 * 16 + l) * DIM_FEAT;

    for (int k = 0; k < DIM_FEAT; k += 4) {
        const int ka = k + hl * 2;
        // A 16x4 f32: lanes 0-15 -> K = k,k+1 ; lanes 16-31 -> K = k+2,k+3
        v2f a  = *(const v2f*)(arow + ka);
        // B 4x16 f32: B[k][n] = W[n][k] (features @ W.T)
        v2f b0 = *(const v2f*)(w0 + ka);
        v2f b1 = *(const v2f*)(w1 + ka);
        v2f b2 = *(const v2f*)(w2 + ka);
        v2f b3 = *(const v2f*)(w3 + ka);
        acc[0] = __builtin_amdgcn_wmma_f32_16x16x4_f32(false, a, false, b0, (short)0, acc[0], false, false);
        acc[1] = __builtin_amdgcn_wmma_f32_16x16x4_f32(false, a, false, b1, (short)0, acc[1], false, false);
        acc[2] = __builtin_amdgcn_wmma_f32_16x16x4_f32(false, a, false, b2, (short)0, acc[2], false, false);
        acc[3] = __builtin_amdgcn_wmma_f32_16x16x4_f32(false, a, false, b3, (short)0, acc[3], false, false);
    }

    // C/D layout: VGPR v -> row v (lanes 0-15) / row v+8 (lanes 16-31); lane -> col
#pragma unroll
    for (int t = 0; t < 4; ++t) {
        const float bn = bias[t * 16 + l];
#pragma unroll
        for (int v = 0; v < 8; ++v) {
            const int m = v + hl * 8;
            float val = acc[t][v] + bn;
            val = val > 0.0f ? val : 0.01f * val;   // leaky_relu 0.01
            out[(size_t)(rowBase + m) * DIM_C + t * 16 + l] = val;
        }
    }
}

// ---------------------------------------------------------------------------
// Row-wise L2 normalize (wave per 64-float row): dst = src / max(||src||, eps)
// ---------------------------------------------------------------------------
__global__ __launch_bounds__(256) void k_normalize_rows(
    const float* __restrict__ src, float* __restrict__ dst, int nrows) {
    const int row  = blockIdx.x * (blockDim.x >> 5) + (threadIdx.x >> 5);
    const int lane = threadIdx.x & 31;
    if (row >= nrows) return;
    v2f v = *(const v2f*)(src + (size_t)row * DIM_C + lane * 2);
    const float ss = wave_reduce_sum(v.x * v.x + v.y * v.y);
    const float inv = 1.0f / fmaxf(sqrtf(ss), 1e-12f);
    v.x *= inv; v.y *= inv;
    *(v2f*)(dst + (size_t)row * DIM_C + lane * 2) = v;
}

// ---------------------------------------------------------------------------
// Per-conv reset: upd = 0, segment max keys = key(-inf), segment sums = 0
// ---------------------------------------------------------------------------
__global__ void k_conv_init(unsigned* __restrict__ mkey, float* __restrict__ s,
                            float* __restrict__ upd, int n_nodes) {
    const int i = blockIdx.x * blockDim.x + threadIdx.x;
    if (i < n_nodes * DIM_C) upd[i] = 0.0f;
    if (i < n_nodes) { mkey[i] = KEY_NEG_INF; s[i] = 0.0f; }
}

// ---------------------------------------------------------------------------
// Phase 1: logits[e] = dot(x[dst], x[src]); segment max over dst (atomic key)
// e >= E handles the flipped half of the final (bidirected) edge set.
// ---------------------------------------------------------------------------
__global__ __launch_bounds__(256) void k_edge_logits(
    const float* __restrict__ x, const int* __restrict__ src,
    const int* __restrict__ dst, int E, int total,
    float* __restrict__ logits, unsigned* __restrict__ mkey) {
    const int e = blockIdx.x * (blockDim.x >> 5) + (threadIdx.x >> 5);
    const int lane = threadIdx.x & 31;
    if (e >= total) return;
    int sI, dI;
    if (e < E) { sI = src[e];     dI = dst[e]; }
    else       { sI = dst[e - E]; dI = src[e - E]; }
    const v2f xi = *(const v2f*)(x + (size_t)dI * DIM_C + lane * 2);
    const v2f xj = *(const v2f*)(x + (size_t)sI * DIM_C + lane * 2);
    const float p = wave_reduce_sum(xi.x * xj.x + xi.y * xj.y);
    if (lane == 0) {
        logits[e] = p;
        atomicMax(&mkey[dI], fkey(p));
    }
}

// ---------------------------------------------------------------------------
// Phase 2: e = exp(logit - m[dst]) (in place over logits); s[dst] += e
// ---------------------------------------------------------------------------
__global__ void k_edge_exp(const int* __restrict__ src, const int* __restrict__ dst,
                           int E, int total, const unsigned* __restrict__ mkey,
                           float* __restrict__ ev, float* __restrict__ s) {
    const int e = blockIdx.x * blockDim.x + threadIdx.x;
    if (e >= total) return;
    const int dI = (e < E) ? dst[e] : src[e - E];
    const float m = ikey(mkey[dI]);
    const float v = expf(ev[e] - m);
    ev[e] = v;
    atomicAdd(&s[dI], v);
}

// ---------------------------------------------------------------------------
// Phase 3: alpha = e/(s[dst]+eps); upd[dst] += alpha * x[src]; optional alpha out
// ---------------------------------------------------------------------------
__global__ __launch_bounds__(256) void k_edge_scatter(
    const float* __restrict__ x, const int* __restrict__ src,
    const int* __restrict__ dst, int E, int total,
    const float* __restrict__ ev, const float* __restrict__ s,
    float* __restrict__ upd, float* __restrict__ alpha_out) {
    const int e = blockIdx.x * (blockDim.x >> 5) + (threadIdx.x >> 5);
    const int lane = threadIdx.x & 31;
    if (e >= total) return;
    int sI, dI;
    if (e < E) { sI = src[e];     dI = dst[e]; }
    else       { sI = dst[e - E]; dI = src[e - E]; }
    const float alpha = ev[e] / (s[dI] + 1e-16f);
    const v2f xj = *(const v2f*)(x + (size_t)sI * DIM_C + lane * 2);
    float* u = upd + (size_t)dI * DIM_C + lane * 2;
    atomicAdd(u,     alpha * xj.x);
    atomicAdd(u + 1, alpha * xj.y);
    if (alpha_out != nullptr && lane == 0) alpha_out[e] = alpha;
}

// ---------------------------------------------------------------------------
// Routing tail: pref = normalize(pref + upd[:NUM_USER]), in place in x
// ---------------------------------------------------------------------------
__global__ __launch_bounds__(256) void k_pref_update(
    float* __restrict__ x, const float* __restrict__ upd) {
    const int row  = blockIdx.x * (blockDim.x >> 5) + (threadIdx.x >> 5);
    const int lane = threadIdx.x & 31;
    if (row >= NUM_USER) return;
    v2f v = *(const v2f*)(x + (size_t)row * DIM_C + lane * 2);
    const v2f u = *(const v2f*)(upd + (size_t)row * DIM_C + lane * 2);
    v.x += u.x; v.y += u.y;
    const float ss = wave_reduce_sum(v.x * v.x + v.y * v.y);
    const float inv = 1.0f / fmaxf(sqrtf(ss), 1e-12f);
    v.x *= inv; v.y *= inv;
    *(v2f*)(x + (size_t)row * DIM_C + lane * 2) = v;
}

// ---------------------------------------------------------------------------
// out = x + leaky_relu(x_hat)
// ---------------------------------------------------------------------------
__global__ void k_final_add(const float* __restrict__ x, const float* __restrict__ upd,
                            float* __restrict__ out, int n) {
    const int i = blockIdx.x * blockDim.x + threadIdx.x;
    if (i >= n) return;
    float h = upd[i];
    h = h > 0.0f ? h : 0.01f * h;
    out[i] = x[i] + h;
}

extern "C" void kernel_launch(void* const* d_in, const int* in_sizes, int n_in,
                              void* d_out, int out_size, void* d_ws, size_t ws_size,
                              hipStream_t stream) {
    const float* features   = (const float*)d_in[0];   // [50000,384]
    const float* preference = (const float*)d_in[1];   // [100000,64]
    const float* W          = (const float*)d_in[2];   // [64,384]
    const float* bias       = (const float*)d_in[3];   // [64]
    const int*   edge       = (const int*)d_in[4];     // [2,1e6]
    const int*   src = edge;                           // item nodes (pre-offset)
    const int*   dst = edge + NUM_EDGES;               // user nodes

    float* out       = (float*)d_out;                          // [150000,64]
    float* out_alpha = out + (size_t)N_NODES * DIM_C;          // [2e6]

    // workspace layout
    float*    x      = (float*)d_ws;                           // [N,64]
    float*    upd    = x + (size_t)N_NODES * DIM_C;            // [N,64]
    float*    logits = upd + (size_t)N_NODES * DIM_C;          // [2E]
    float*    s      = logits + 2 * (size_t)NUM_EDGES;         // [N]
    unsigned* mkey   = (unsigned*)(s + N_NODES);               // [N]
    float*    xfeat  = x + (size_t)NUM_USER * DIM_C;

    const dim3 blk(256);
    const int wavesPerBlk = 8;

    // feats = leaky_relu(features @ W.T + b)  (WMMA f32)
    {
        const int tiles = NUM_ITEM / 16;
        k_gemm_feats<<<dim3((tiles + wavesPerBlk - 1) / wavesPerBlk), blk, 0, stream>>>(
            features, W, bias, xfeat);
    }
    // normalize feats in place; normalize preference into x's user region
    k_normalize_rows<<<dim3((NUM_ITEM + wavesPerBlk - 1) / wavesPerBlk), blk, 0, stream>>>(
        xfeat, xfeat, NUM_ITEM);
    k_normalize_rows<<<dim3((NUM_USER + wavesPerBlk - 1) / wavesPerBlk), blk, 0, stream>>>(
        preference, x, NUM_USER);

    const int initBlocks = (N_NODES * DIM_C + 255) / 256;

    // 3 routing iterations (edges: item -> user)
    for (int r = 0; r < 3; ++r) {
        k_conv_init<<<dim3(initBlocks), blk, 0, stream>>>(mkey, s, upd, N_NODES);
        k_edge_logits<<<dim3((NUM_EDGES + wavesPerBlk - 1) / wavesPerBlk), blk, 0, stream>>>(
            x, src, dst, NUM_EDGES, NUM_EDGES, logits, mkey);
        k_edge_exp<<<dim3((NUM_EDGES + 255) / 256), blk, 0, stream>>>(
            src, dst, NUM_EDGES, NUM_EDGES, mkey, logits, s);
        k_edge_scatter<<<dim3((NUM_EDGES + wavesPerBlk - 1) / wavesPerBlk), blk, 0, stream>>>(
            x, src, dst, NUM_EDGES, NUM_EDGES, logits, s, upd, nullptr);
        k_pref_update<<<dim3((NUM_USER + wavesPerBlk - 1) / wavesPerBlk), blk, 0, stream>>>(
            x, upd);
    }

    // final bidirected conv (2E edges), alpha emitted
    const int T = 2 * NUM_EDGES;
    k_conv_init<<<dim3(initBlocks), blk, 0, stream>>>(mkey, s, upd, N_NODES);
    k_edge_logits<<<dim3((T + wavesPerBlk - 1) / wavesPerBlk), blk, 0, stream>>>(
        x, src, dst, NUM_EDGES, T, logits, mkey);
    k_edge_exp<<<dim3((T + 255) / 256), blk, 0, stream>>>(
        src, dst, NUM_EDGES, T, mkey, logits, s);
    k_edge_scatter<<<dim3((T + wavesPerBlk - 1) / wavesPerBlk), blk, 0, stream>>>(
        x, src, dst, NUM_EDGES, T, logits, s, upd, out_alpha);

    k_final_add<<<dim3(initBlocks), blk, 0, stream>>>(x, upd, out, N_NODES * DIM_C);
}